// GCNConvLayer_36361193128559
// MI455X (gfx1250) — compile-verified
//
#include <hip/hip_runtime.h>

#define N_NODES 50000
#define N_EDGES 800000
#define DIM     128

typedef __attribute__((ext_vector_type(2))) float v2f;
typedef __attribute__((ext_vector_type(8))) float v8f;

// ---------- phase 1: degrees (self-loop folded in as deg=1 init) ----------
__global__ void k_init_deg(float* __restrict__ deg) {
    int i = blockIdx.x * blockDim.x + threadIdx.x;
    if (i < N_NODES) deg[i] = 1.0f;
}

__global__ void k_count_deg(const int* __restrict__ ei, float* __restrict__ deg) {
    int e = blockIdx.x * blockDim.x + threadIdx.x;
    if (e < N_EDGES) {
        int dst = ei[N_EDGES + e];
        unsafeAtomicAdd(&deg[dst], 1.0f);   // counts are small -> exact in f32
    }
}

__global__ void k_rsqrt(float* __restrict__ deg) {
    int i = blockIdx.x * blockDim.x + threadIdx.x;
    if (i < N_NODES) deg[i] = rsqrtf(deg[i]);   // deg >= 1 always
}

// ---------- phase 2: acc = x * dinv^2 (self-loop message, atomic-free) ----------
__global__ void k_init_acc(const float* __restrict__ x, const float* __restrict__ dinv,
                           float* __restrict__ acc) {
    int g = blockIdx.x * blockDim.x + threadIdx.x;   // over N*D/4 float4s
    if (g < N_NODES * (DIM / 4)) {
        int n = g >> 5;                               // DIM/4 == 32
        float s = dinv[n]; s = s * s;
        float4 v = ((const float4*)x)[g];
        v.x *= s; v.y *= s; v.z *= s; v.w *= s;
        ((float4*)acc)[g] = v;
    }
}

// ---------- phase 3: one wave per edge; lane handles a float4 of features ----------
__global__ void k_scatter(const float* __restrict__ x, const int* __restrict__ ei,
                          const float* __restrict__ dinv, float* __restrict__ acc) {
    int g = blockIdx.x * blockDim.x + threadIdx.x;
    int e = g >> 5;          // wave id == edge id
    int lane = g & 31;
    if (e < N_EDGES) {
        int src = ei[e];
        int dst = ei[N_EDGES + e];
        float nrm = dinv[src] * dinv[dst];
        float4 v = ((const float4*)(x + (size_t)src * DIM))[lane];
        float* p = acc + (size_t)dst * DIM + lane * 4;
        unsafeAtomicAdd(p + 0, v.x * nrm);
        unsafeAtomicAdd(p + 1, v.y * nrm);
        unsafeAtomicAdd(p + 2, v.z * nrm);
        unsafeAtomicAdd(p + 3, v.w * nrm);
    }
}

// ---------- phase 4: out = x + relu(acc @ W + b), f32 WMMA 16x16x4 ----------
// block = 256 threads = 8 waves; block computes 16 rows x 128 cols.
// Wave w computes columns [16w, 16w+16). LDS holds the 16x128 acc tile with a
// 132-float row stride (132 mod 64 = 4) so A-frag ds_load_b64s are bank-conflict-free.
__global__ void __launch_bounds__(256) k_gemm_fused(
        const float* __restrict__ acc, const float* __restrict__ x,
        const float* __restrict__ W, const float* __restrict__ bias,
        float* __restrict__ out) {
    __shared__ float At[16 * 132];
    const int tid  = threadIdx.x;
    const int wave = tid >> 5;
    const int lane = tid & 31;
    const int row0 = blockIdx.x * 16;

    // cooperative tile load: 16 rows x 32 float4
    {
        const float4* src4 = (const float4*)(acc + (size_t)row0 * DIM);
        float4* dst4 = (float4*)At;
        #pragma unroll
        for (int i = tid; i < 16 * 32; i += 256) {
            int r = i >> 5, q = i & 31;
            dst4[r * 33 + q] = src4[i];          // row stride 33 float4 = 132 floats
        }
    }
    __syncthreads();

    const int col0 = wave * 16;
    const int nn   = lane & 15;                  // N (and M) index within tile
    const int koff = (lane >> 4) << 1;           // lanes 16-31 handle K+2,K+3
    const int colg = col0 + nn;

    // preload B fragments: bfrag[kk] = { W[4kk+koff][colg], W[4kk+koff+1][colg] }
    v2f bfrag[32];
    #pragma unroll
    for (int kk = 0; kk < 32; ++kk) {
        int k = (kk << 2) + koff;
        v2f t;
        t.x = W[k * DIM + colg];
        t.y = W[(k + 1) * DIM + colg];
        bfrag[kk] = t;
    }

    v8f c = {};
    const float* arow = At + nn * 132;
    #pragma unroll
    for (int kk = 0; kk < 32; ++kk) {
        int k = (kk << 2) + koff;
        v2f a;
        a.x = arow[k];
        a.y = arow[k + 1];
        // D = A(16x4,f32) * B(4x16,f32) + C(16x16,f32)
        c = __builtin_amdgcn_wmma_f32_16x16x4_f32(false, a, false, bfrag[kk],
                                                  (short)0, c, false, false);
    }

    // epilogue: bias + relu + residual, C layout: VGPR r -> row r (+8 for lanes>=16)
    const float bv = bias[colg];
    const int rbase = row0 + ((lane >> 4) << 3);
    #pragma unroll
    for (int r = 0; r < 8; ++r) {
        size_t off = (size_t)(rbase + r) * DIM + colg;
        float v = c[r] + bv;
        v = v > 0.0f ? v : 0.0f;
        out[off] = x[off] + v;
    }
}

extern "C" void kernel_launch(void* const* d_in, const int* in_sizes, int n_in,
                              void* d_out, int out_size, void* d_ws, size_t ws_size,
                              hipStream_t stream) {
    (void)in_sizes; (void)n_in; (void)out_size; (void)ws_size;
    const float* x  = (const float*)d_in[0];
    const int*   ei = (const int*)d_in[1];     // [2, E] int32
    const float* W  = (const float*)d_in[2];
    const float* b  = (const float*)d_in[3];
    float* out = (float*)d_out;

    char* ws  = (char*)d_ws;
    float* deg = (float*)ws;                                        // N floats (becomes dinv)
    float* acc = (float*)(ws + ((N_NODES * 4 + 255) & ~255));       // N*D floats (~25.6 MB)

    k_init_deg <<<(N_NODES + 255) / 256, 256, 0, stream>>>(deg);
    k_count_deg<<<(N_EDGES + 255) / 256, 256, 0, stream>>>(ei, deg);
    k_rsqrt    <<<(N_NODES + 255) / 256, 256, 0, stream>>>(deg);
    k_init_acc <<<(N_NODES * (DIM / 4) + 255) / 256, 256, 0, stream>>>(x, deg, acc);
    k_scatter  <<<(N_EDGES * 32 + 255) / 256, 256, 0, stream>>>(x, ei, deg, acc);
    k_gemm_fused<<<N_NODES / 16, 256, 0, stream>>>(acc, x, W, b, out);
}